// ConvTransformer_89446988907152
// MI455X (gfx1250) — compile-verified
//
#include <hip/hip_runtime.h>
#include <hip/hip_bf16.h>

// IBP through conv2d(C_IN=8 -> C_OUT=16, 3x3, pad 1, 28x28), recast as GEMM:
//   L[16,784] = Wp @ im2col(l) + Wn @ im2col(u) + b
//   U[16,784] = Wp @ im2col(u) + Wn @ im2col(l) + b
// Computed with V_WMMA_F32_16X16X4_F32: M=16 (C_OUT), K=72 (18 steps of 4),
// N=784 (49 tiles of 16). One wave32 per N-tile.

typedef __attribute__((ext_vector_type(2))) float v2f;
typedef __attribute__((ext_vector_type(8))) float v8f;

#define C_IN   8
#define C_OUT  16
#define HW     28
#define NPIX   (HW * HW)        // 784
#define KTOT   (C_IN * 9)       // 72
#define KSTEPS (KTOT / 4)       // 18
#define N_OUT  (C_OUT * NPIX)   // 12544
#define NTILES (NPIX / 16)      // 49

// Fetch one im2col element pair-source: row k, column (ho,wo) of im2col(lb/ub).
// Branchless zero-padding: clamp address (always valid), select 0 when OOB,
// so EXEC remains all-1s (required by WMMA).
__device__ __forceinline__ void im2col_pair(const float* __restrict__ lb,
                                            const float* __restrict__ ub,
                                            int k, int ho, int wo,
                                            float& bl, float& bu) {
    int ci = k / 9;
    int r  = k - ci * 9;
    int kh = r / 3;
    int kw = r - kh * 3;
    int ih = ho + kh - 1;
    int iw = wo + kw - 1;
    bool ok = (ih >= 0) && (ih < HW) && (iw >= 0) && (iw < HW);
    int ihc = min(max(ih, 0), HW - 1);
    int iwc = min(max(iw, 0), HW - 1);
    int idx = ci * NPIX + ihc * HW + iwc;
    float l = lb[idx];
    float u = ub[idx];
    bl = ok ? l : 0.0f;
    bu = ok ? u : 0.0f;
}

__global__ __launch_bounds__(32)
void ibp_conv_wmma_f32(const float* __restrict__ lb,   // [6272]
                       const float* __restrict__ ub,   // [6272]
                       const float* __restrict__ kern, // [16*8*3*3] = [16][72]
                       const float* __restrict__ bias, // [16]
                       float* __restrict__ out)        // [2*12544]: lower | upper
{
    const int lane  = threadIdx.x;        // 0..31
    const int tile  = blockIdx.x;         // 0..48
    const int lo16  = lane & 15;          // A: row m=co ; B: col n ; D: col n
    const int khalf = lane >> 4;          // selects K-pair within a K=4 step

    const int n  = tile * 16 + lo16;      // output spatial position 0..783
    const int ho = n / HW;
    const int wo = n - ho * HW;

    v8f accL = {};   // lower-bound accumulator (16x16 f32 tile)
    v8f accU = {};   // upper-bound accumulator

#pragma unroll
    for (int s = 0; s < KSTEPS; ++s) {
        const int k0 = 4 * s + 2 * khalf; // this lane's first K of the step

        // ---- A fragments: Wp / Wn, row co = lo16, cols k0, k0+1 ----
        const float w0 = kern[lo16 * KTOT + k0];
        const float w1 = kern[lo16 * KTOT + k0 + 1];
        v2f ap, an;
        ap.x = fmaxf(w0, 0.0f);  ap.y = fmaxf(w1, 0.0f);
        an.x = fminf(w0, 0.0f);  an.y = fminf(w1, 0.0f);

        // ---- B fragments: im2col(l) / im2col(u), rows k0,k0+1, col n ----
        v2f bl, bu;
        {
            float l0, u0;
            im2col_pair(lb, ub, k0, ho, wo, l0, u0);
            bl.x = l0;  bu.x = u0;
        }
        {
            float l1, u1;
            im2col_pair(lb, ub, k0 + 1, ho, wo, l1, u1);
            bl.y = l1;  bu.y = u1;
        }

        // ---- accumulate the four interval-GEMM terms ----
        accL = __builtin_amdgcn_wmma_f32_16x16x4_f32(false, ap, false, bl,
                                                     (short)0, accL, false, false);
        accL = __builtin_amdgcn_wmma_f32_16x16x4_f32(false, an, false, bu,
                                                     (short)0, accL, false, false);
        accU = __builtin_amdgcn_wmma_f32_16x16x4_f32(false, ap, false, bu,
                                                     (short)0, accU, false, false);
        accU = __builtin_amdgcn_wmma_f32_16x16x4_f32(false, an, false, bl,
                                                     (short)0, accU, false, false);
    }

    // D layout (32-bit 16x16): VGPR r holds M=r (lanes 0-15) / M=r+8 (lanes 16-31),
    // lane%16 = N. Add per-channel bias and store lower | upper.
#pragma unroll
    for (int r = 0; r < 8; ++r) {
        const int m  = r + 8 * khalf;     // output channel
        const float bv = bias[m];
        out[m * NPIX + n]         = accL[r] + bv;
        out[N_OUT + m * NPIX + n] = accU[r] + bv;
    }
}

extern "C" void kernel_launch(void* const* d_in, const int* in_sizes, int n_in,
                              void* d_out, int out_size, void* d_ws, size_t ws_size,
                              hipStream_t stream) {
    (void)in_sizes; (void)n_in; (void)out_size; (void)d_ws; (void)ws_size;
    const float* lb   = (const float*)d_in[0];  // lower_bound_prev [1,6272,1]
    const float* ub   = (const float*)d_in[1];  // upper_bound_prev [1,6272,1]
    const float* kern = (const float*)d_in[2];  // kernel [16,8,3,3]
    const float* bias = (const float*)d_in[3];  // bias [16]
    float* out = (float*)d_out;                 // lower|upper, 2*12544 floats

    ibp_conv_wmma_f32<<<NTILES, 32, 0, stream>>>(lb, ub, kern, bias, out);
}